// ConditioningGNN_86543591014924
// MI455X (gfx1250) — compile-verified
//
#include <hip/hip_runtime.h>
#include <hip/hip_bf16.h>

typedef float v2f __attribute__((ext_vector_type(2)));
typedef float v8f __attribute__((ext_vector_type(8)));
typedef float f4  __attribute__((ext_vector_type(4)));

#define DMODEL 512
#define NHEAD  8
#define DHEAD  64
#define NCTX   64
#define LN_EPS 1e-5f

// ---------------------------------------------------------------------------
// Kernel A: Q/K/V projections.
//  blocks 0..31 : 256 waves total, each wave computes one 16x16 tile of
//                 KH = ctx @ Wk^T + bk  or  VH = ctx @ Wv^T + bv
//                 using V_WMMA_F32_16X16X4_F32 over K=512 (128 wmma ops).
//  block 32     : qh = path_token @ Wq^T + bq (scalar, tiny).
// ---------------------------------------------------------------------------
__global__ void __launch_bounds__(256)
cgnn_qkv_kernel(const float* __restrict__ H,
                const float* __restrict__ path_token,
                const float* __restrict__ W_in1,
                const float* __restrict__ b_in1,
                const int*   __restrict__ context_ids,
                float* __restrict__ qh,
                float* __restrict__ KH,
                float* __restrict__ VH)
{
    if (blockIdx.x == 32) {
        int t = threadIdx.x;
        for (int n = t; n < DMODEL; n += 256) {
            const float* w = W_in1 + (size_t)n * DMODEL;   // Wq row n
            float acc = b_in1[n];
            for (int k = 0; k < DMODEL; ++k) acc += path_token[k] * w[k];
            qh[n] = acc;
        }
        return;
    }

    const int wave = (blockIdx.x << 3) + (threadIdx.x >> 5);  // 0..255
    const int lane = threadIdx.x & 31;
    const int hl   = lane >> 4;        // 0: K pair {0,1}, 1: K pair {2,3}
    const int lr   = lane & 15;        // row within tile

    const int matrix = wave >> 7;      // 0 -> K (Wk), 1 -> V (Wv)
    const int tile   = wave & 127;
    const int m0     = (tile >> 5) << 4;   // 0,16,32,48   (ctx rows)
    const int n0     = (tile & 31) << 4;   // 0..496       (output features)

    const float* W    = W_in1 + (size_t)(matrix + 1) * DMODEL * DMODEL;
    const float* bias = b_in1 + (matrix + 1) * DMODEL;

    // A fragment: ctx rows (gathered from H); lane lr -> row m0+lr
    const int arow = context_ids[m0 + lr];
    const float* Arow = H + (size_t)arow * DMODEL + 2 * hl;
    // B fragment: B[k][n] = W[n][k]  -> lane lr reads row n0+lr of W
    const float* Brow = W + (size_t)(n0 + lr) * DMODEL + 2 * hl;

    // C init with bias (depends only on n = n0 + lane%16)
    const float bv = bias[n0 + lr];
    v8f c;
#pragma unroll
    for (int i = 0; i < 8; ++i) c[i] = bv;

    for (int k = 0; k < DMODEL; k += 4) {
        v2f a = *(const v2f*)(Arow + k);
        v2f b = *(const v2f*)(Brow + k);
        c = __builtin_amdgcn_wmma_f32_16x16x4_f32(
                /*neg_a=*/false, a, /*neg_b=*/false, b,
                /*c_mod=*/(short)0, c, /*reuse_a=*/false, /*reuse_b=*/false);
    }

    // D layout: VGPR i, lanes 0-15: (M=i, N=lane); lanes 16-31: (M=i+8, N=lane-16)
    float* OUT = matrix ? VH : KH;
#pragma unroll
    for (int i = 0; i < 8; ++i)
        OUT[(size_t)(m0 + i + 8 * hl) * DMODEL + n0 + lr] = c[i];
}

// ---------------------------------------------------------------------------
// Kernel B: single workgroup serial tail:
//   scores -> softmax -> attn out -> out-proj(+residual) -> LN1 -> Wv2 -> W_out2
// Produces attn2[512] in workspace.
// ---------------------------------------------------------------------------
__global__ void __launch_bounds__(256)
cgnn_attn_kernel(const float* __restrict__ path_token,
                 const float* __restrict__ W_out1, const float* __restrict__ b_out1,
                 const float* __restrict__ W_in2,  const float* __restrict__ b_in2,
                 const float* __restrict__ W_out2, const float* __restrict__ b_out2,
                 const float* __restrict__ ln1_g,  const float* __restrict__ ln1_b,
                 const float* __restrict__ qh,
                 const float* __restrict__ KH,
                 const float* __restrict__ VH,
                 float* __restrict__ attn2)
{
    __shared__ float s_q[DMODEL];
    __shared__ float s_sc[DMODEL];     // scores, then softmax weights [h*64+k]
    __shared__ float s_out[DMODEL];
    __shared__ float s_upd[DMODEL];
    __shared__ float s_v[DMODEL];
    __shared__ float s_stat[2 * NHEAD];
    __shared__ float s_red[16];
    __shared__ float s_mu, s_rstd;

    const int t = threadIdx.x;

    s_q[t]       = qh[t];
    s_q[t + 256] = qh[t + 256];
    __syncthreads();

    // scores[h][k] = <q_h, k_{k,h}> / sqrt(64)
    for (int idx = t; idx < DMODEL; idx += 256) {
        int h = idx >> 6, k = idx & 63;
        const float* kr = KH + (size_t)k * DMODEL + h * DHEAD;
        const float* qr = s_q + h * DHEAD;
        float s = 0.f;
        for (int d = 0; d < DHEAD; ++d) s += qr[d] * kr[d];
        s_sc[idx] = s * 0.125f;
    }
    __syncthreads();

    if (t < NHEAD) {
        float m = -1e30f;
        for (int k = 0; k < NCTX; ++k) m = fmaxf(m, s_sc[t * 64 + k]);
        float sum = 0.f;
        for (int k = 0; k < NCTX; ++k) sum += __expf(s_sc[t * 64 + k] - m);
        s_stat[2 * t] = m;
        s_stat[2 * t + 1] = sum;
    }
    __syncthreads();

    for (int idx = t; idx < DMODEL; idx += 256) {
        int h = idx >> 6;
        s_sc[idx] = __expf(s_sc[idx] - s_stat[2 * h]) / s_stat[2 * h + 1];
    }
    __syncthreads();

    // out[h*64+d] = sum_k w[h][k] * VH[k][h*64+d]
    for (int idx = t; idx < DMODEL; idx += 256) {
        int h = idx >> 6, d = idx & 63;
        const float* wrow = s_sc + h * 64;
        float o = 0.f;
        for (int k = 0; k < NCTX; ++k) o += wrow[k] * VH[(size_t)k * DMODEL + h * DHEAD + d];
        s_out[idx] = o;
    }
    __syncthreads();

    // out projection + residual, accumulate LN stats
    float part = 0.f, partsq = 0.f;
    for (int n = t; n < DMODEL; n += 256) {
        const float* w = W_out1 + (size_t)n * DMODEL;
        float acc = b_out1[n];
        for (int m = 0; m < DMODEL; ++m) acc += s_out[m] * w[m];
        acc += path_token[n];
        s_upd[n] = acc;
        part += acc;
        partsq += acc * acc;
    }
#pragma unroll
    for (int off = 16; off > 0; off >>= 1) {
        part   += __shfl_xor(part, off, 32);
        partsq += __shfl_xor(partsq, off, 32);
    }
    if ((t & 31) == 0) { s_red[t >> 5] = part; s_red[8 + (t >> 5)] = partsq; }
    __syncthreads();
    if (t == 0) {
        float s = 0.f, sq = 0.f;
        for (int i = 0; i < 8; ++i) { s += s_red[i]; sq += s_red[8 + i]; }
        float mu = s * (1.f / DMODEL);
        float var = sq * (1.f / DMODEL) - mu * mu;
        s_mu = mu;
        s_rstd = rsqrtf(var + LN_EPS);
    }
    __syncthreads();

    for (int n = t; n < DMODEL; n += 256)
        s_upd[n] = (s_upd[n] - s_mu) * s_rstd * ln1_g[n] + ln1_b[n];
    __syncthreads();

    // v = upd @ Wv2^T + bv2
    const float* Wv2 = W_in2 + (size_t)2 * DMODEL * DMODEL;
    const float* bv2 = b_in2 + 2 * DMODEL;
    for (int n = t; n < DMODEL; n += 256) {
        const float* w = Wv2 + (size_t)n * DMODEL;
        float acc = bv2[n];
        for (int m = 0; m < DMODEL; ++m) acc += s_upd[m] * w[m];
        s_v[n] = acc;
    }
    __syncthreads();

    // attn2 = v @ W_out2^T + b_out2
    for (int n = t; n < DMODEL; n += 256) {
        const float* w = W_out2 + (size_t)n * DMODEL;
        float acc = b_out2[n];
        for (int m = 0; m < DMODEL; ++m) acc += s_v[m] * w[m];
        attn2[n] = acc;
    }
}

// ---------------------------------------------------------------------------
// Kernel C: H_cond = layer_norm(attn2 + H) over N rows. Bandwidth-bound.
// One wave per row: 32 lanes x 16 floats; NT loads/stores; shfl reduction.
// ---------------------------------------------------------------------------
__global__ void __launch_bounds__(256)
cgnn_lnbcast_kernel(const float* __restrict__ H,
                    const float* __restrict__ attn2,
                    const float* __restrict__ g,
                    const float* __restrict__ b,
                    float* __restrict__ out, int N)
{
    const int wid  = threadIdx.x >> 5;
    const int lane = threadIdx.x & 31;
    const int row  = blockIdx.x * 8 + wid;
    if (row >= N) return;

    const f4* hr = (const f4*)(H   + (size_t)row * DMODEL + lane * 16);
    f4*       orw = (f4*)(out + (size_t)row * DMODEL + lane * 16);
    const f4* a4 = (const f4*)(attn2 + lane * 16);
    const f4* g4 = (const f4*)(g + lane * 16);
    const f4* b4 = (const f4*)(b + lane * 16);

    f4 x[4];
    float s = 0.f, sq = 0.f;
#pragma unroll
    for (int j = 0; j < 4; ++j) {
        f4 hv = __builtin_nontemporal_load(hr + j);
        f4 v  = hv + a4[j];
        x[j] = v;
        s  += v.x + v.y + v.z + v.w;
        sq += v.x * v.x + v.y * v.y + v.z * v.z + v.w * v.w;
    }
#pragma unroll
    for (int off = 16; off > 0; off >>= 1) {
        s  += __shfl_xor(s,  off, 32);
        sq += __shfl_xor(sq, off, 32);
    }
    const float mu   = s * (1.f / DMODEL);
    const float var  = sq * (1.f / DMODEL) - mu * mu;
    const float rstd = rsqrtf(var + LN_EPS);
#pragma unroll
    for (int j = 0; j < 4; ++j) {
        f4 v = (x[j] - mu) * rstd * g4[j] + b4[j];
        __builtin_nontemporal_store(v, orw + j);
    }
}

// ---------------------------------------------------------------------------
extern "C" void kernel_launch(void* const* d_in, const int* in_sizes, int n_in,
                              void* d_out, int out_size, void* d_ws, size_t ws_size,
                              hipStream_t stream)
{
    const float* H          = (const float*)d_in[0];
    const float* path_token = (const float*)d_in[1];
    const float* W_in1      = (const float*)d_in[2];
    const float* b_in1      = (const float*)d_in[3];
    const float* W_out1     = (const float*)d_in[4];
    const float* b_out1     = (const float*)d_in[5];
    const float* W_in2      = (const float*)d_in[6];
    const float* b_in2      = (const float*)d_in[7];
    const float* W_out2     = (const float*)d_in[8];
    const float* b_out2     = (const float*)d_in[9];
    const float* ln1_g      = (const float*)d_in[10];
    const float* ln1_b      = (const float*)d_in[11];
    const float* ln2_g      = (const float*)d_in[12];
    const float* ln2_b      = (const float*)d_in[13];
    /* d_in[14] = edge_index : unused by the reference */
    const int*   context_ids = (const int*)d_in[15];

    const int N = in_sizes[0] / DMODEL;

    float* ws    = (float*)d_ws;
    float* qh    = ws;                       // 512
    float* KH    = qh + DMODEL;              // 64*512
    float* VH    = KH + NCTX * DMODEL;       // 64*512
    float* attn2 = VH + NCTX * DMODEL;       // 512

    cgnn_qkv_kernel<<<33, 256, 0, stream>>>(H, path_token, W_in1, b_in1,
                                            context_ids, qh, KH, VH);
    cgnn_attn_kernel<<<1, 256, 0, stream>>>(path_token, W_out1, b_out1,
                                            W_in2, b_in2, W_out2, b_out2,
                                            ln1_g, ln1_b, qh, KH, VH, attn2);
    cgnn_lnbcast_kernel<<<(N + 7) / 8, 256, 0, stream>>>(H, attn2, ln2_g, ln2_b,
                                                         (float*)d_out, N);
}